// vPFAE_ClusterGCN_63462436766187
// MI455X (gfx1250) — compile-verified
//
#include <hip/hip_runtime.h>
#include <hip/hip_bf16.h>

typedef __attribute__((ext_vector_type(2))) float v2f;
typedef __attribute__((ext_vector_type(8))) float v8f;

// ---------------------------------------------------------------------------
// Degree / normalization precompute  (deg includes self loop -> deg >= 1)
// ---------------------------------------------------------------------------
__global__ void vgae_deg_init(float* __restrict__ deg, int n) {
    int i = blockIdx.x * blockDim.x + threadIdx.x;
    if (i < n) deg[i] = 1.0f;              // self loop
}

__global__ void vgae_deg_count(const int* __restrict__ dst, float* __restrict__ deg, int e) {
    int i = blockIdx.x * blockDim.x + threadIdx.x;
    if (i < e) atomicAdd(&deg[dst[i]], 1.0f);
}

__global__ void vgae_norms(const float* __restrict__ deg, float* __restrict__ dinv,
                           float* __restrict__ dis, int n) {
    int i = blockIdx.x * blockDim.x + threadIdx.x;
    if (i < n) {
        float d = deg[i];
        dinv[i] = 1.0f / d;      // deg >= 1 always
        dis[i]  = rsqrtf(d);
    }
}

// ---------------------------------------------------------------------------
// ClusterGCN aggregation:  agg = D^-1 (A+I) x
//   init:    agg[i] = x[i] * dinv[i]           (self loop)
//   scatter: agg[dst] += x[src] * dinv[dst]    (edges, float4 chunks)
// ---------------------------------------------------------------------------
template <int F>
__global__ void vgae_agg_init(const float* __restrict__ x, const float* __restrict__ dinv,
                              float* __restrict__ agg, int n) {
    constexpr int C = F / 4;
    long idx = (long)blockIdx.x * blockDim.x + threadIdx.x;
    if (idx >= (long)n * C) return;
    int node = (int)(idx / C);
    float w = dinv[node];
    float4 v = ((const float4*)x)[idx];
    float4 r = make_float4(v.x * w, v.y * w, v.z * w, v.w * w);
    ((float4*)agg)[idx] = r;
}

template <int F>
__global__ void vgae_agg_scatter(const float* __restrict__ x, const int* __restrict__ src,
                                 const int* __restrict__ dst, const float* __restrict__ dinv,
                                 float* __restrict__ agg, int e) {
    constexpr int C = F / 4;
    long idx = (long)blockIdx.x * blockDim.x + threadIdx.x;
    if (idx >= (long)e * C) return;
    int ed = (int)(idx / C);
    int c  = (int)(idx % C);
    int s = src[ed], d = dst[ed];
    float w = dinv[d];
    float4 v = ((const float4*)x)[(long)s * C + c];
    float* p = agg + (long)d * F + c * 4;
    atomicAdd(p + 0, v.x * w);
    atomicAdd(p + 1, v.y * w);
    atomicAdd(p + 2, v.z * w);
    atomicAdd(p + 3, v.w * w);
}

// ---------------------------------------------------------------------------
// Dual GEMM with fp32 WMMA:  out = relu(A1 @ B1 + A2 @ B2 + bias)
// One wave = one 16-row tile; block = 4 waves = 64 rows.
// B1/B2 transposed into LDS as [OUT][IN] so B-fragment loads are ds_load_b64.
// V_WMMA_F32_16X16X4_F32 lane mapping (wave32):
//   A: lane holds rows m=lane&15, K pair k0+2*(lane>>4)   -> v2f
//   B: lane holds col  n=lane&15, K pair k0+2*(lane>>4)   -> v2f
//   C: vgpr r holds (row r + 8*(lane>>4), col lane&15)    -> v8f
// ---------------------------------------------------------------------------
template <int IN, int OUT>
__global__ void vgae_gemm_cluster(const float* __restrict__ A1, const float* __restrict__ A2,
                                  const float* __restrict__ B1g, const float* __restrict__ B2g,
                                  const float* __restrict__ bias, float* __restrict__ out, int n) {
    constexpr int NT = OUT / 16;
    extern __shared__ float sm[];
    float* w1 = sm;              // [OUT][IN]
    float* w2 = sm + OUT * IN;   // [OUT][IN]

    int tid = threadIdx.x;
    for (int i = tid; i < IN * OUT; i += blockDim.x) {
        int k = i / OUT, o = i % OUT;      // W is row-major (IN, OUT)
        w1[o * IN + k] = B1g[i];
        w2[o * IN + k] = B2g[i];
    }
    __syncthreads();

    int lane = tid & 31;
    int wave = tid >> 5;
    int base = blockIdx.x * 64 + wave * 16;
    int m  = lane & 15;          // A row / B col within tile
    int kh = (lane >> 4) * 2;    // K pair offset
    int mh = (lane >> 4) * 8;    // C row offset

    int rowc = base + m;
    if (rowc > n - 1) rowc = n - 1;        // clamp loads; EXEC stays all-1s
    long rowL = (long)rowc;

    v8f c[NT];
#pragma unroll
    for (int t = 0; t < NT; ++t) {
        float bv = bias[t * 16 + m];
        c[t] = (v8f){bv, bv, bv, bv, bv, bv, bv, bv};
    }

    const float* a1p = A1 + rowL * IN + kh;
    const float* a2p = A2 + rowL * IN + kh;
    for (int k0 = 0; k0 < IN; k0 += 4) {
        v2f a1 = *(const v2f*)(a1p + k0);
        v2f a2 = *(const v2f*)(a2p + k0);
#pragma unroll
        for (int t = 0; t < NT; ++t) {
            v2f bb1 = *(const v2f*)(&w1[(t * 16 + m) * IN + k0 + kh]);
            v2f bb2 = *(const v2f*)(&w2[(t * 16 + m) * IN + k0 + kh]);
            c[t] = __builtin_amdgcn_wmma_f32_16x16x4_f32(false, a1, false, bb1,
                                                         (short)0, c[t], false, false);
            c[t] = __builtin_amdgcn_wmma_f32_16x16x4_f32(false, a2, false, bb2,
                                                         (short)0, c[t], false, false);
        }
    }

#pragma unroll
    for (int t = 0; t < NT; ++t) {
#pragma unroll
        for (int r = 0; r < 8; ++r) {
            int row = base + r + mh;
            if (row < n) {
                float v = fmaxf(c[t][r], 0.0f);          // ReLU
                out[(long)row * OUT + t * 16 + m] = v;
            }
        }
    }
}

// ---------------------------------------------------------------------------
// Head GEMMs (shared A):  hm = h2 @ Wmu,  hl = h2 @ Wls   (no bias, no relu)
// ---------------------------------------------------------------------------
__global__ void vgae_gemm_heads(const float* __restrict__ A, const float* __restrict__ Bmg,
                                const float* __restrict__ Blg, float* __restrict__ om,
                                float* __restrict__ ol, int n) {
    constexpr int IN = 64, OUT = 32, NT = 2;
    extern __shared__ float sm[];
    float* wm = sm;
    float* wl = sm + OUT * IN;

    int tid = threadIdx.x;
    for (int i = tid; i < IN * OUT; i += blockDim.x) {
        int k = i / OUT, o = i % OUT;
        wm[o * IN + k] = Bmg[i];
        wl[o * IN + k] = Blg[i];
    }
    __syncthreads();

    int lane = tid & 31;
    int wave = tid >> 5;
    int base = blockIdx.x * 64 + wave * 16;
    int m  = lane & 15;
    int kh = (lane >> 4) * 2;
    int mh = (lane >> 4) * 8;

    int rowc = base + m;
    if (rowc > n - 1) rowc = n - 1;
    long rowL = (long)rowc;

    v8f cm[NT], cl[NT];
#pragma unroll
    for (int t = 0; t < NT; ++t) {
        cm[t] = (v8f){0.f, 0.f, 0.f, 0.f, 0.f, 0.f, 0.f, 0.f};
        cl[t] = (v8f){0.f, 0.f, 0.f, 0.f, 0.f, 0.f, 0.f, 0.f};
    }

    const float* ap = A + rowL * IN + kh;
    for (int k0 = 0; k0 < IN; k0 += 4) {
        v2f a = *(const v2f*)(ap + k0);
#pragma unroll
        for (int t = 0; t < NT; ++t) {
            v2f bm = *(const v2f*)(&wm[(t * 16 + m) * IN + k0 + kh]);
            v2f bl = *(const v2f*)(&wl[(t * 16 + m) * IN + k0 + kh]);
            cm[t] = __builtin_amdgcn_wmma_f32_16x16x4_f32(false, a, false, bm,
                                                          (short)0, cm[t], false, false);
            cl[t] = __builtin_amdgcn_wmma_f32_16x16x4_f32(false, a, false, bl,
                                                          (short)0, cl[t], false, false);
        }
    }

#pragma unroll
    for (int t = 0; t < NT; ++t) {
#pragma unroll
        for (int r = 0; r < 8; ++r) {
            int row = base + r + mh;
            if (row < n) {
                om[(long)row * OUT + t * 16 + m] = cm[t][r];
                ol[(long)row * OUT + t * 16 + m] = cl[t][r];
            }
        }
    }
}

// ---------------------------------------------------------------------------
// GCN head aggregation:  out = D^-1/2 (A+I) D^-1/2 h + b
//   init:    out[i] = b + h[i] * dis[i]^2
//   scatter: out[dst] += h[src] * dis[src]*dis[dst]
// ---------------------------------------------------------------------------
__global__ void vgae_head_init(const float* __restrict__ hm, const float* __restrict__ hl,
                               const float* __restrict__ dis, const float* __restrict__ bmu,
                               const float* __restrict__ bls, float* __restrict__ om,
                               float* __restrict__ ol, int n) {
    long idx = (long)blockIdx.x * blockDim.x + threadIdx.x;
    if (idx >= (long)n * 8) return;
    int node = (int)(idx >> 3);
    int c = (int)(idx & 7);
    float d = dis[node];
    float d2 = d * d;
    float4 bm = ((const float4*)bmu)[c];
    float4 bl = ((const float4*)bls)[c];
    float4 vm = ((const float4*)hm)[idx];
    float4 vl = ((const float4*)hl)[idx];
    ((float4*)om)[idx] = make_float4(vm.x * d2 + bm.x, vm.y * d2 + bm.y,
                                     vm.z * d2 + bm.z, vm.w * d2 + bm.w);
    ((float4*)ol)[idx] = make_float4(vl.x * d2 + bl.x, vl.y * d2 + bl.y,
                                     vl.z * d2 + bl.z, vl.w * d2 + bl.w);
}

__global__ void vgae_head_scatter(const float* __restrict__ hm, const float* __restrict__ hl,
                                  const int* __restrict__ src, const int* __restrict__ dst,
                                  const float* __restrict__ dis, float* __restrict__ om,
                                  float* __restrict__ ol, int e) {
    long idx = (long)blockIdx.x * blockDim.x + threadIdx.x;
    if (idx >= (long)e * 8) return;
    int ed = (int)(idx >> 3);
    int c = (int)(idx & 7);
    int s = src[ed], d = dst[ed];
    float w = dis[s] * dis[d];
    float4 vm = ((const float4*)hm)[(long)s * 8 + c];
    float4 vl = ((const float4*)hl)[(long)s * 8 + c];
    float* pm = om + (long)d * 32 + c * 4;
    atomicAdd(pm + 0, vm.x * w);
    atomicAdd(pm + 1, vm.y * w);
    atomicAdd(pm + 2, vm.z * w);
    atomicAdd(pm + 3, vm.w * w);
    float* pl = ol + (long)d * 32 + c * 4;
    atomicAdd(pl + 0, vl.x * w);
    atomicAdd(pl + 1, vl.y * w);
    atomicAdd(pl + 2, vl.z * w);
    atomicAdd(pl + 3, vl.w * w);
}

// ---------------------------------------------------------------------------
// Orchestration
// ---------------------------------------------------------------------------
extern "C" void kernel_launch(void* const* d_in, const int* in_sizes, int n_in,
                              void* d_out, int out_size, void* d_ws, size_t ws_size,
                              hipStream_t stream) {
    const float* x   = (const float*)d_in[0];
    const int*   ei  = (const int*)d_in[1];
    const float* W1  = (const float*)d_in[2];
    const float* b1  = (const float*)d_in[3];
    const float* Wr1 = (const float*)d_in[4];
    const float* W2  = (const float*)d_in[5];
    const float* b2  = (const float*)d_in[6];
    const float* Wr2 = (const float*)d_in[7];
    const float* Wmu = (const float*)d_in[8];
    const float* bmu = (const float*)d_in[9];
    const float* Wls = (const float*)d_in[10];
    const float* bls = (const float*)d_in[11];

    const int IN = 128, L1d = 96, L2d = 64, OUTd = 32;
    const int N = in_sizes[0] / IN;
    const int E = in_sizes[1] / 2;
    const int* src = ei;
    const int* dst = ei + E;

    float* ws   = (float*)d_ws;
    float* deg  = ws;                               // N
    float* dinv = deg + N;                          // N
    float* dis  = dinv + N;                         // N
    float* agg  = dis + N;                          // N*128 (reused as hm/hl)
    float* h1   = agg + (size_t)N * IN;             // N*96
    float* h2   = h1 + (size_t)N * L1d;             // N*64
    float* hm   = agg;                              // N*32 (agg free after layer 2)
    float* hl   = agg + (size_t)N * OUTd;           // N*32

    float* om = (float*)d_out;                      // mu:     N*32
    float* ol = om + (size_t)N * OUTd;              // logstd: N*32

    auto cdiv = [](long a, long b) { return (int)((a + b - 1) / b); };

    // degrees & normalizations (shared by all layers)
    vgae_deg_init<<<cdiv(N, 256), 256, 0, stream>>>(deg, N);
    vgae_deg_count<<<cdiv(E, 256), 256, 0, stream>>>(dst, deg, E);
    vgae_norms<<<cdiv(N, 256), 256, 0, stream>>>(deg, dinv, dis, N);

    // layer 1: h1 = relu(agg @ W1 + b1 + x @ Wr1)
    vgae_agg_init<128><<<cdiv((long)N * 32, 256), 256, 0, stream>>>(x, dinv, agg, N);
    vgae_agg_scatter<128><<<cdiv((long)E * 32, 256), 256, 0, stream>>>(x, src, dst, dinv, agg, E);
    vgae_gemm_cluster<128, 96><<<cdiv(N, 64), 128, 2 * 128 * 96 * sizeof(float), stream>>>(
        agg, x, W1, Wr1, b1, h1, N);

    // layer 2: h2 = relu(agg2 @ W2 + b2 + h1 @ Wr2)
    vgae_agg_init<96><<<cdiv((long)N * 24, 256), 256, 0, stream>>>(h1, dinv, agg, N);
    vgae_agg_scatter<96><<<cdiv((long)E * 24, 256), 256, 0, stream>>>(h1, src, dst, dinv, agg, E);
    vgae_gemm_cluster<96, 64><<<cdiv(N, 64), 128, 2 * 96 * 64 * sizeof(float), stream>>>(
        agg, h1, W2, Wr2, b2, h2, N);

    // heads: hm = h2 @ Wmu, hl = h2 @ Wls; then GCN-normalized aggregation + bias
    vgae_gemm_heads<<<cdiv(N, 64), 128, 2 * 64 * 32 * sizeof(float), stream>>>(
        h2, Wmu, Wls, hm, hl, N);
    vgae_head_init<<<cdiv((long)N * 8, 256), 256, 0, stream>>>(hm, hl, dis, bmu, bls, om, ol, N);
    vgae_head_scatter<<<cdiv((long)E * 8, 256), 256, 0, stream>>>(hm, hl, src, dst, dis, om, ol, E);
}